// EdgeConv_13692355739964
// MI455X (gfx1250) — compile-verified
//
#include <hip/hip_runtime.h>

typedef float v2f __attribute__((ext_vector_type(2)));
typedef float v8f __attribute__((ext_vector_type(8)));
typedef int   v4i_vs __attribute__((vector_size(4 * sizeof(int))));

#define NN      50000   // vertices
#define CIN     128
#define COUT    128
#define K2      256     // 2*CIN
#define TILE_E  48      // edges per workgroup tile (3 subtiles of 16 per wave)
#define XSTRIDE 260     // padded row stride (floats) -> bank-conflict free
#define NEG_SLOPE 0.3f
#define NBLOCKS 2048    // persistent workgroups

#if __has_builtin(__builtin_amdgcn_global_load_async_to_lds_b128)
#define ASYNC_GATHER 1
#else
#define ASYNC_GATHER 0
#endif

__device__ __forceinline__ void wait_async_zero() {
#if __has_builtin(__builtin_amdgcn_s_wait_asynccnt)
    __builtin_amdgcn_s_wait_asynccnt(0);
#else
    asm volatile("s_wait_asynccnt 0x0" ::: "memory");
#endif
}

// ---------------- kernel 0: zero the scatter accumulators ----------------
__global__ void zero_ws_kernel(float* __restrict__ sums, float* __restrict__ counts) {
    int tid = blockIdx.x * blockDim.x + threadIdx.x;
    if (tid < NN * COUT) sums[tid] = 0.0f;
    if (tid < NN)        counts[tid] = 0.0f;
}

// ------------- kernel 0b: fold edge-subtract into weights -------------
// W' = [W1 - W2 ; W2] so that W'.[x_i ; x_j] == W.[x_i ; x_j - x_i]
__global__ void wprep_kernel(const float* __restrict__ w, float* __restrict__ wp) {
    int tid = blockIdx.x * blockDim.x + threadIdx.x;
    if (tid >= COUT * K2) return;
    int k = tid & (K2 - 1);
    float v = w[tid];
    if (k < CIN) v -= w[tid + CIN];
    wp[tid] = v;
}

// ---------------- kernel 1: transpose in[128][N] -> inT[N][128] ----------------
__global__ void transpose_kernel(const float* __restrict__ in, float* __restrict__ out) {
    __shared__ float tile[32][33];
    int x  = blockIdx.x * 32 + threadIdx.x;   // node index
    int y0 = blockIdx.y * 32;                 // channel base
#pragma unroll
    for (int j = 0; j < 32; j += 8) {
        int c = y0 + threadIdx.y + j;
        tile[threadIdx.y + j][threadIdx.x] = (x < NN) ? in[c * NN + x] : 0.0f;
    }
    __syncthreads();
    int c = y0 + threadIdx.x;
#pragma unroll
    for (int j = 0; j < 32; j += 8) {
        int n = blockIdx.x * 32 + threadIdx.y + j;
        if (n < NN) out[n * CIN + c] = tile[threadIdx.x][threadIdx.y + j];
    }
}

// ------- kernel 2: double-buffered async gather -> WMMA GEMM -> scatter-add -------
__global__ void __launch_bounds__(256)
edgeconv_main_kernel(const float* __restrict__ inT, const float* __restrict__ wp,
                     const int* __restrict__ ridx, const int* __restrict__ gidx,
                     float* __restrict__ sums, float* __restrict__ counts,
                     int E, int ntiles)
{
    __shared__ float Xs[2][TILE_E * XSTRIDE];   // 2 x 49,920 B
    __shared__ int   nodeIds[2][TILE_E];

    const int tid  = threadIdx.x;
    const int wave = tid >> 5;
    const int lane = tid & 31;
    const int half = lane >> 4;     // 0 or 1
    const int l16  = lane & 15;
    const int mb   = wave * 16;     // 8 waves cover 128 output channels
    const int koff = half * 2;      // lanes 16-31 hold K=2,3 of each 4-wide k-step

    // gather for tile t into buffer buf: 12 async ops per wave, always issued
    // (out-of-range edges clamp to E-1; their columns are discarded at scatter)
    auto issue_gather = [&](int t, int buf) {
        const int eb = t * TILE_E;
#pragma unroll
        for (int i = 0; i < 6; ++i) {            // 6 edges per wave * 8 waves = 48
            int el = wave * 6 + i;
            int e  = eb + el;
            int ec = (e < E) ? e : (E - 1);
            int r = ridx[ec];
            int g = gidx[ec];
            float* xr = &Xs[buf][el * XSTRIDE];
            const float* grp = inT + r * CIN + 4 * lane;
            const float* ggp = inT + g * CIN + 4 * lane;
#if ASYNC_GATHER
            __builtin_amdgcn_global_load_async_to_lds_b128(
                (__attribute__((address_space(1))) v4i_vs*)grp,
                (__attribute__((address_space(3))) v4i_vs*)(xr + 4 * lane), 0, 0);
            __builtin_amdgcn_global_load_async_to_lds_b128(
                (__attribute__((address_space(1))) v4i_vs*)ggp,
                (__attribute__((address_space(3))) v4i_vs*)(xr + CIN + 4 * lane), 0, 0);
#else
            *(float4*)(xr + 4 * lane)       = *(const float4*)grp;
            *(float4*)(xr + CIN + 4 * lane) = *(const float4*)ggp;
#endif
            if (lane == 0) {
                if (e < E) {
                    nodeIds[buf][el] = r;
                    unsafeAtomicAdd(&counts[r], 1.0f);
                } else {
                    nodeIds[buf][el] = -1;
                }
            }
        }
    };

    // ---- prologue: start DMA for first tile, then load W' strip into registers ----
    const int t0 = blockIdx.x;
    if (t0 < ntiles) issue_gather(t0, 0);

    const float* wrow = wp + (mb + l16) * K2 + koff;
    v2f wf[64];
#pragma unroll
    for (int kk = 0; kk < 64; ++kk)
        wf[kk] = *(const v2f*)(wrow + kk * 4);

    int ibuf = 0;
    for (int t = t0; t < ntiles; t += gridDim.x) {
        // ---- A: wait for current buffer's DMA (only its 12 ops outstanding) ----
#if ASYNC_GATHER
        wait_async_zero();
#endif
        // ---- B: all waves' writes to Xs[ibuf] complete & visible ----
        __syncthreads();

        // ---- C: kick off next tile's DMA into the other buffer ----
        int tn = t + gridDim.x;
        if (tn < ntiles) issue_gather(tn, 1 - ibuf);

        // ---- D: GEMM 16x48 per wave, K=256 via V_WMMA_F32_16X16X4_F32 ----
        const float* xb    = &Xs[ibuf][0];
        const float* xrow0 = xb + ( 0 + l16) * XSTRIDE + koff;
        const float* xrow1 = xb + (16 + l16) * XSTRIDE + koff;
        const float* xrow2 = xb + (32 + l16) * XSTRIDE + koff;
        v8f acc[3] = {};
#pragma unroll
        for (int kk = 0; kk < 64; ++kk) {
            v2f a  = wf[kk];
            v2f b0 = *(const v2f*)(xrow0 + kk * 4);
            v2f b1 = *(const v2f*)(xrow1 + kk * 4);
            v2f b2 = *(const v2f*)(xrow2 + kk * 4);
            acc[0] = __builtin_amdgcn_wmma_f32_16x16x4_f32(false, a, false, b0,
                                                           (short)0, acc[0], false, false);
            acc[1] = __builtin_amdgcn_wmma_f32_16x16x4_f32(false, a, false, b1,
                                                           (short)0, acc[1], false, false);
            acc[2] = __builtin_amdgcn_wmma_f32_16x16x4_f32(false, a, false, b2,
                                                           (short)0, acc[2], false, false);
        }

        // ---- E: scatter-add (lane's column = edge l16 of subtile s) ----
        const int eb = t * TILE_E;
#pragma unroll
        for (int s = 0; s < 3; ++s) {
            int e = eb + s * 16 + l16;
            if (e < E) {
                int node = nodeIds[ibuf][s * 16 + l16];
                float* sp = sums + node * COUT + mb + half * 8;
#pragma unroll
                for (int i = 0; i < 8; ++i)
                    unsafeAtomicAdd(sp + i, acc[s][i]);
            }
        }
        ibuf ^= 1;
    }
}

// ---------------- kernel 3: mean + bias + LeakyReLU, write [C_OUT][N] ----------------
__global__ void finalize_kernel(const float* __restrict__ sums, const float* __restrict__ counts,
                                const float* __restrict__ bias, float* __restrict__ out)
{
    int tid = blockIdx.x * blockDim.x + threadIdx.x;
    if (tid >= NN * COUT) return;
    int c = tid / NN;
    int n = tid - c * NN;
    float cnt = counts[n];
    float v = sums[n * COUT + c] / fmaxf(cnt, 1.0f);
    if (cnt > 0.0f) v += bias[c];
    out[tid] = (v >= 0.0f) ? v : NEG_SLOPE * v;
}

extern "C" void kernel_launch(void* const* d_in, const int* in_sizes, int n_in,
                              void* d_out, int out_size, void* d_ws, size_t ws_size,
                              hipStream_t stream)
{
    const float* in_features = (const float*)d_in[0];   // [1,128,50000] f32
    const float* weight      = (const float*)d_in[1];   // [128,256] f32
    const float* bias        = (const float*)d_in[2];   // [128] f32
    const int*   ridx        = (const int*)d_in[3];     // [E] int32
    const int*   gidx        = (const int*)d_in[4];     // [E] int32
    float*       out         = (float*)d_out;           // [128,50000] f32
    const int    E           = in_sizes[3];

    // workspace: inT (25.6MB) | sums (25.6MB) | counts (0.2MB) | W' (128KB)
    float* inT    = (float*)d_ws;
    float* sums   = inT  + (size_t)NN * CIN;
    float* counts = sums + (size_t)NN * COUT;
    float* wprime = counts + NN;

    zero_ws_kernel<<<(NN * COUT + 255) / 256, 256, 0, stream>>>(sums, counts);
    wprep_kernel<<<(COUT * K2 + 255) / 256, 256, 0, stream>>>(weight, wprime);

    dim3 tb(32, 8);
    dim3 tg((NN + 31) / 32, CIN / 32);
    transpose_kernel<<<tg, tb, 0, stream>>>(in_features, inT);

    int ntiles  = (E + TILE_E - 1) / TILE_E;
    int nblocks = (ntiles < NBLOCKS) ? ntiles : NBLOCKS;
    edgeconv_main_kernel<<<nblocks, 256, 0, stream>>>(inT, wprime, ridx, gidx,
                                                      sums, counts, E, ntiles);

    finalize_kernel<<<(NN * COUT + 255) / 256, 256, 0, stream>>>(sums, counts, bias, out);
}